// BilevelFramework_41669772706056
// MI455X (gfx1250) — compile-verified
//
#include <hip/hip_runtime.h>
#include <math.h>

typedef __attribute__((ext_vector_type(2))) float v2f;
typedef __attribute__((ext_vector_type(8))) float v8f;

// ---------------------------------------------------------------------------
// Kernel 1: E[i,j] = (adj>0 && i!=j) ? exp(-lam*dist[i,j]) : 0 ; also Et = E^T
// ---------------------------------------------------------------------------
__global__ __launch_bounds__(256) void build_E_kernel(
    const int* __restrict__ adj, const float* __restrict__ dist,
    const int* __restrict__ lamPtr, float* __restrict__ E,
    float* __restrict__ Et, int n) {
  int idx = blockIdx.x * blockDim.x + threadIdx.x;
  int n2 = n * n;
  if (idx >= n2) return;
  int i = idx / n;
  int j = idx - i * n;
  float lam = (float)(*lamPtr);
  float e = 0.0f;
  if (adj[idx] > 0 && i != j) e = expf(-lam * dist[idx]);
  E[idx] = e;
  Et[(size_t)j * n + i] = e;
}

// ---------------------------------------------------------------------------
// Kernel 2: f32 WMMA GEMM, C = A*B (accumulate==0) or C += A*B (accumulate!=0)
// A, B, C row-major n x n, n % 16 == 0. One wave per 16x16 C tile,
// V_WMMA_F32_16X16X4_F32 along K.
// A frag:  lane l holds A[tM+(l&15), k + 2*(l>>4) + {0,1}]        (float2)
// B frag:  lane l holds B[k + 2*(l>>4) + {0,1}, tN+(l&15)]        (2 scalars)
// C/D:     vgpr r -> C[tM + r + 8*(l>>4), tN + (l&15)]
// ---------------------------------------------------------------------------
__global__ __launch_bounds__(256) void gemm_f32_wmma_kernel(
    const float* __restrict__ A, const float* __restrict__ B,
    float* __restrict__ C, int n, int nTiles, int accumulate) {
  const int wave = threadIdx.x >> 5;
  const int lane = threadIdx.x & 31;
  const int tile = blockIdx.x * 8 + wave;
  const int totalTiles = nTiles * nTiles;
  if (tile >= totalTiles) return;  // wave-uniform; EXEC stays all-ones

  const int tM = (tile / nTiles) * 16;
  const int tN = (tile % nTiles) * 16;
  const int half = lane >> 4;   // 0: K pair {0,1}, 1: K pair {2,3}
  const int l15  = lane & 15;

  const float* aRow = A + (size_t)(tM + l15) * n;  // + k
  const float* bCol = B + (tN + l15);              // + k*n

  v8f acc = {};
  if (accumulate) {
#pragma unroll
    for (int r = 0; r < 8; ++r)
      acc[r] = C[(size_t)(tM + r + half * 8) * n + tN + l15];
  }

  for (int k = 0; k < n; k += 8) {
    const int ka = k + half * 2;
    v2f a0 = *(const v2f*)(aRow + ka);
    v2f b0;
    b0.x = bCol[(size_t)ka * n];
    b0.y = bCol[(size_t)(ka + 1) * n];
    v2f a1 = *(const v2f*)(aRow + ka + 4);
    v2f b1;
    b1.x = bCol[(size_t)(ka + 4) * n];
    b1.y = bCol[(size_t)(ka + 5) * n];
    // 8 args: (neg_a, A, neg_b, B, c_mod, C, reuse_a, reuse_b)
    acc = __builtin_amdgcn_wmma_f32_16x16x4_f32(false, a0, false, b0,
                                                (short)0, acc, false, false);
    acc = __builtin_amdgcn_wmma_f32_16x16x4_f32(false, a1, false, b1,
                                                (short)0, acc, false, false);
  }

#pragma unroll
  for (int r = 0; r < 8; ++r)
    C[(size_t)(tM + r + half * 8) * n + tN + l15] = acc[r];
}

// ---------------------------------------------------------------------------
// Kernel 3: W = demand / (E + S) where positive; write W into S-slot AND out.
// ---------------------------------------------------------------------------
__global__ __launch_bounds__(256) void weights_kernel(
    const float* __restrict__ od, const float* __restrict__ E,
    float* __restrict__ Wslot /* holds S on entry */,
    float* __restrict__ out, int n) {
  int idx = blockIdx.x * blockDim.x + threadIdx.x;
  int n2 = n * n;
  if (idx >= n2) return;
  int i = idx / n;
  int j = idx - i * n;
  float t = E[idx] + Wslot[idx];          // E + S
  float o = od[idx];
  float dem = (o > 0.0f && i != j) ? o : 0.0f;
  float w = (t > 0.0f) ? dem / t : 0.0f;
  Wslot[idx] = w;
  out[idx] = w;                            // direct-path seed: out = W
}

// ---------------------------------------------------------------------------
// Kernel 4: flows = E .* (W + P + Q) ; out already holds W+P+Q.
// ---------------------------------------------------------------------------
__global__ __launch_bounds__(256) void finalize_kernel(
    const float* __restrict__ E, float* __restrict__ out, int n2) {
  int idx = blockIdx.x * blockDim.x + threadIdx.x;
  if (idx >= n2) return;
  out[idx] = E[idx] * out[idx];
}

extern "C" void kernel_launch(void* const* d_in, const int* in_sizes, int n_in,
                              void* d_out, int out_size, void* d_ws,
                              size_t ws_size, hipStream_t stream) {
  const float* od   = (const float*)d_in[0];
  const int*   adj  = (const int*)d_in[1];
  const float* dist = (const float*)d_in[2];
  const int*   lam  = (const int*)d_in[3];
  (void)d_in[4];  // capacity unused by the returned flows

  const int n  = (int)(sqrtf((float)in_sizes[0]) + 0.5f);  // 384
  const int n2 = n * n;

  float* E     = (float*)d_ws;   // n2 floats
  float* Et    = E + n2;         // n2 floats
  float* Wslot = Et + n2;        // n2 floats: S, then W
  float* out   = (float*)d_out;

  const int eltBlocks = (n2 + 255) / 256;
  const int nTiles = n / 16;
  const int gemmBlocks = (nTiles * nTiles + 7) / 8;  // 8 waves per block

  // 1. E, E^T
  build_E_kernel<<<eltBlocks, 256, 0, stream>>>(adj, dist, lam, E, Et, n);
  // 2. S = E @ E  (into Wslot)
  gemm_f32_wmma_kernel<<<gemmBlocks, 256, 0, stream>>>(E, E, Wslot, n, nTiles, 0);
  // 3. W = demand/(E+S); out = W
  weights_kernel<<<eltBlocks, 256, 0, stream>>>(od, E, Wslot, out, n);
  // 4. out += W @ E^T   (first-hop factor)
  gemm_f32_wmma_kernel<<<gemmBlocks, 256, 0, stream>>>(Wslot, Et, out, n, nTiles, 1);
  // 5. out += E^T @ W   (second-hop factor)
  gemm_f32_wmma_kernel<<<gemmBlocks, 256, 0, stream>>>(Et, Wslot, out, n, nTiles, 1);
  // 6. flows = E .* out
  finalize_kernel<<<eltBlocks, 256, 0, stream>>>(E, out, n2);
}